// PPLMLATrainRotaryPositionEmbeddingLayer_61521111548175
// MI455X (gfx1250) — compile-verified
//
#include <hip/hip_runtime.h>
#include <cstddef>

// Problem constants (from reference): BSZ=1, SEQ=2048, H=128
// q last dim = 192 (128 nope + 64 rope), kv last dim = 256 (128 nope + 128 v)
#define SEQ 2048
#define H   128
#define DQ  192
#define DKV 256
#define DR  64
#define DN  128

typedef float f32x4 __attribute__((ext_vector_type(4)));

// ---- gfx1250 async-to-LDS path (guarded; fallback = plain loads) ----------
#if defined(__has_builtin)
#if __has_builtin(__builtin_amdgcn_global_load_async_to_lds_b64)
#define HAVE_ASYNC_LDS 1
#endif
#endif
#ifndef HAVE_ASYNC_LDS
#define HAVE_ASYNC_LDS 0
#endif

#if defined(__has_builtin)
#if __has_builtin(__builtin_amdgcn_s_wait_asynccnt)
#define WAIT_ASYNC0() __builtin_amdgcn_s_wait_asynccnt(0)
#else
#define WAIT_ASYNC0() asm volatile("s_wait_asynccnt 0" ::: "memory")
#endif
#else
#define WAIT_ASYNC0() asm volatile("s_wait_asynccnt 0" ::: "memory")
#endif

#if HAVE_ASYNC_LDS
// b64 payload type per clang diagnostic: int __attribute__((vector_size(8)))
typedef int v2i __attribute__((vector_size(8)));
typedef __attribute__((address_space(1))) v2i gv2i_t;   // global (prints as __device__)
typedef __attribute__((address_space(3))) v2i lv2i_t;   // LDS
// Generic LDS pointer -> 32-bit LDS offset pointer (generic LDS addr = {SHARED_BASE, off[31:0]})
__device__ __forceinline__ lv2i_t* lds_cast64(const void* p) {
  return (lv2i_t*)(unsigned)(unsigned long long)p;
}
__device__ __forceinline__ gv2i_t* glob_cast64(const void* p) {
  return (gv2i_t*)(unsigned long long)p;
}
#endif

__global__ __launch_bounds__(256) void mla_rope_fused(
    const float* __restrict__ q,      // [SEQ][H][192]
    const float* __restrict__ kv,     // [SEQ][H][256]
    const float* __restrict__ k_pe,   // [SEQ][64]
    const float* __restrict__ cos_t,  // [SEQ][64]
    const float* __restrict__ sin_t,  // [SEQ][64]
    const int*   __restrict__ pos,    // [SEQ]
    float* __restrict__ q_out,        // [SEQ][H][192]
    float* __restrict__ res_kv)       // [SEQ][2][H][192]
{
  const int s = blockIdx.x;
  const int t = threadIdx.x;

  __shared__ __align__(16) float sc[DR];     // cos row
  __shared__ __align__(16) float ss[DR];     // sin row
  __shared__ __align__(16) float skpe[DR];   // raw k_pe row
  __shared__ __align__(16) float skrot[DR];  // rope(k_pe) row

  const int p = pos[s];

  // Prefetch the kv row (consumed by the second loop) into the hierarchy.
  {
    const char* pf = (const char*)(kv + (size_t)s * H * DKV);
#pragma unroll
    for (int i = 0; i < 4; ++i)
      __builtin_prefetch(pf + (size_t)(t + i * 256) * 128, 0, 0);
  }

  // ---- Stage cos/sin/k_pe rows (256 B each) into LDS --------------------
#if HAVE_ASYNC_LDS
  if (t < 32) {            // wave 0: cos row (32 lanes x 8 B = 256 B)
    __builtin_amdgcn_global_load_async_to_lds_b64(
        glob_cast64(cos_t + (size_t)p * DR) + t, lds_cast64(sc) + t, 0, 0);
  } else if (t < 64) {     // wave 1: sin row
    const int l = t - 32;
    __builtin_amdgcn_global_load_async_to_lds_b64(
        glob_cast64(sin_t + (size_t)p * DR) + l, lds_cast64(ss) + l, 0, 0);
  } else if (t < 96) {     // wave 2: k_pe row
    const int l = t - 64;
    __builtin_amdgcn_global_load_async_to_lds_b64(
        glob_cast64(k_pe + (size_t)s * DR) + l, lds_cast64(skpe) + l, 0, 0);
  }
  WAIT_ASYNC0();
  __syncthreads();
#else
  if (t < DR) {
    sc[t]   = cos_t[(size_t)p * DR + t];
    ss[t]   = sin_t[(size_t)p * DR + t];
    skpe[t] = k_pe[(size_t)s * DR + t];
  }
  __syncthreads();
#endif

  // rope(k_pe) computed once per block, broadcast to all 128 heads.
  if (t < DR) {
    const float x  = skpe[t];
    const float xr = (t < 32) ? -skpe[t + 32] : skpe[t - 32];
    skrot[t] = x * sc[t] + xr * ss[t];
  }
  __syncthreads();

  const f32x4* c4  = (const f32x4*)sc;
  const f32x4* s4  = (const f32x4*)ss;
  const f32x4* kr4 = (const f32x4*)skrot;

  // ---- Loop 1: q_out = [q_nope | rope(q_pe)] ----------------------------
  {
    const f32x4* qb = (const f32x4*)(q + (size_t)s * H * DQ);   // 48 f32x4 / head
    f32x4* ob = (f32x4*)(q_out + (size_t)s * H * DQ);
    for (int f = t; f < H * 48; f += 256) {
      const int h  = f / 48;
      const int i4 = f - h * 48;
      const f32x4* qrow = qb + h * 48;
      f32x4 v;
      if (i4 < 32) {
        v = qrow[i4];                                  // pure copy (nope)
      } else {
        const int r = i4 - 32;                          // 0..15 within rope half
        const f32x4 x  = qrow[i4];
        const f32x4 xp = qrow[32 + ((r + 8) & 15)];     // rotate-half partner
        const float sg = (r < 8) ? -1.0f : 1.0f;
        v = x * c4[r] + (sg * xp) * s4[r];
      }
      __builtin_nontemporal_store(v, &ob[f]);
    }
  }

  // ---- Loops 2+3: res_kv[s][0] = [k_nope | kpe_rot], res_kv[s][1] = [v | 0]
  {
    const f32x4* kvb = (const f32x4*)(kv + (size_t)s * H * DKV); // 64 f32x4 / head
    f32x4* kb = (f32x4*)(res_kv + (size_t)(s * 2)     * H * DQ);
    f32x4* vb = (f32x4*)(res_kv + (size_t)(s * 2 + 1) * H * DQ);
    const f32x4 z4 = {0.0f, 0.0f, 0.0f, 0.0f};
    for (int f = t; f < H * 48; f += 256) {
      const int h  = f / 48;
      const int i4 = f - h * 48;
      const f32x4* kvrow = kvb + h * 64;
      const f32x4 kvv = (i4 < 32) ? kvrow[i4]      : kr4[i4 - 32];
      const f32x4 vvv = (i4 < 32) ? kvrow[32 + i4] : z4;
      __builtin_nontemporal_store(kvv, &kb[f]);
      __builtin_nontemporal_store(vvv, &vb[f]);
    }
  }
}

extern "C" void kernel_launch(void* const* d_in, const int* in_sizes, int n_in,
                              void* d_out, int out_size, void* d_ws, size_t ws_size,
                              hipStream_t stream) {
  const float* q     = (const float*)d_in[0];
  const float* kv    = (const float*)d_in[1];
  const float* k_pe  = (const float*)d_in[2];
  const float* cos_t = (const float*)d_in[3];
  const float* sin_t = (const float*)d_in[4];
  const int*   pos   = (const int*)d_in[5];

  float* q_out  = (float*)d_out;
  float* res_kv = q_out + (size_t)SEQ * H * DQ;   // outputs concatenated flat

  mla_rope_fused<<<dim3(SEQ), dim3(256), 0, stream>>>(
      q, kv, k_pe, cos_t, sin_t, pos, q_out, res_kv);

  (void)in_sizes; (void)n_in; (void)d_ws; (void)ws_size; (void)out_size;
}